// ClasswiseECE_635655159837
// MI455X (gfx1250) — compile-verified
//
#include <hip/hip_runtime.h>
#include <hip/hip_bf16.h>
#include <stdint.h>

#define NBINS 15
#define CCLS  128
#define NSEG  (CCLS * NBINS)          // 1920 segments
#define WAVES_PER_BLOCK 8
#define BLOCK_THREADS   (WAVES_PER_BLOCK * 32)
#define GRID_BLOCKS     1024
#define DEPTH           8             // async row-pipeline depth per wave

// ---------------- float atomic helpers (native fadd paths) ----------------
__device__ __forceinline__ void atomAddWG(float* p, float v) {
    __hip_atomic_fetch_add(p, v, __ATOMIC_RELAXED, __HIP_MEMORY_SCOPE_WORKGROUP);
}
__device__ __forceinline__ void atomAddDev(float* p, float v) {
    __hip_atomic_fetch_add(p, v, __ATOMIC_RELAXED, __HIP_MEMORY_SCOPE_AGENT);
}

// ---------------- kernel 0: zero workspace ----------------
__global__ void ece_zero_ws(float* ws) {
    int i = blockIdx.x * blockDim.x + threadIdx.x;
    if (i < 3 * NSEG + 1) ws[i] = 0.0f;   // last word doubles as int maxlabel = 0
}

// ---------------- kernel 1: softmax + classwise-bin accumulation ----------------
__global__ __launch_bounds__(BLOCK_THREADS)
void ece_accum(const float* __restrict__ logits,
               const int*   __restrict__ labels,
               float* __restrict__ g_cnt,
               float* __restrict__ g_conf,
               float* __restrict__ g_acc,
               int*   __restrict__ g_maxlab,
               int N)
{
    __shared__ float s_cnt[NSEG];
    __shared__ float s_conf[NSEG];
    __shared__ float s_acc[NSEG];
    __shared__ int   s_maxlab;
    __shared__ __align__(16) float s_buf[WAVES_PER_BLOCK][DEPTH][CCLS];

    const int tid = threadIdx.x;
    for (int i = tid; i < NSEG; i += BLOCK_THREADS) {
        s_cnt[i] = 0.0f; s_conf[i] = 0.0f; s_acc[i] = 0.0f;
    }
    if (tid == 0) s_maxlab = 0;
    __syncthreads();

    const int lane   = tid & 31;
    const int wave   = tid >> 5;
    const int gwave  = blockIdx.x * WAVES_PER_BLOCK + wave;
    const int nwaves = gridDim.x * WAVES_PER_BLOCK;

    // LDS byte offset of this wave's buffer block (low 32 bits of flat shared addr)
    const uint32_t bufBase = (uint32_t)(uintptr_t)(&s_buf[wave][0][0]);

    // async copy of one full row (512B) into an LDS slot: 32 lanes x 16B
    auto prefetch = [&](int r, int slot) {
        const float* src = logits + (size_t)r * (size_t)CCLS + (size_t)(lane << 2);
        uint64_t ga = (uint64_t)(uintptr_t)src;
        uint32_t la = bufBase + (uint32_t)slot * (CCLS * 4u) + (uint32_t)(lane << 4);
        asm volatile("global_load_async_to_lds_b128 %0, %1, off"
                     :: "v"(la), "v"(ga) : "memory");
    };
    // dummy async issue (keeps ASYNCcnt arithmetic uniform; 32B instead of 512B)
    auto dummy = [&](int slot) {
        uint64_t ga = (uint64_t)(uintptr_t)(logits + lane);
        uint32_t la = bufBase + (uint32_t)slot * (CCLS * 4u) + (uint32_t)lane;
        asm volatile("global_load_async_to_lds_b8 %0, %1, off"
                     :: "v"(la), "v"(ga) : "memory");
    };
    auto issue = [&](long long r, int slot) {
        if (r < (long long)N) prefetch((int)r, slot);
        else                  dummy(slot);
    };

    if (gwave < N) {
        #pragma unroll
        for (int t = 0; t < DEPTH - 1; ++t)
            issue((long long)gwave + (long long)t * nwaves, t);

        int k = 0;
        for (int row = gwave; row < N; row += nwaves, ++k) {
            issue((long long)row + (long long)(DEPTH - 1) * nwaves, (k + DEPTH - 1) & (DEPTH - 1));
            asm volatile("s_wait_asynccnt 0x7" ::: "memory"); // oldest (row k) landed in LDS

            const float4 v = *reinterpret_cast<const float4*>(&s_buf[wave][k & (DEPTH - 1)][lane << 2]);
            const int lab = labels[row];

            // row max (wave32 tree)
            float m = fmaxf(fmaxf(v.x, v.y), fmaxf(v.z, v.w));
            #pragma unroll
            for (int d = 16; d >= 1; d >>= 1) m = fmaxf(m, __shfl_xor(m, d, 32));

            float e0 = __expf(v.x - m), e1 = __expf(v.y - m);
            float e2 = __expf(v.z - m), e3 = __expf(v.w - m);
            float s = (e0 + e1) + (e2 + e3);
            #pragma unroll
            for (int d = 16; d >= 1; d >>= 1) s += __shfl_xor(s, d, 32);
            const float inv = 1.0f / s;

            float p[4] = {e0 * inv, e1 * inv, e2 * inv, e3 * inv};
            #pragma unroll
            for (int j = 0; j < 4; ++j) {
                const float pj = p[j];
                if (pj > 0.0f) {                       // valid = p > 0
                    const int c = (lane << 2) + j;
                    int b = (int)ceilf(pj * 15.0f) - 1;
                    b = b < 0 ? 0 : (b > NBINS - 1 ? NBINS - 1 : b);
                    const int seg = c * NBINS + b;
                    atomAddWG(&s_cnt[seg],  1.0f);
                    atomAddWG(&s_conf[seg], pj);
                    if (lab == c) atomAddWG(&s_acc[seg], 1.0f);
                }
            }
            if (lane == 0) atomicMax(&s_maxlab, lab);
        }
        asm volatile("s_wait_asynccnt 0x0" ::: "memory");
    }
    __syncthreads();

    // merge block-private histogram into global workspace
    for (int i = tid; i < NSEG; i += BLOCK_THREADS) {
        float c = s_cnt[i];  if (c != 0.0f) atomAddDev(&g_cnt[i],  c);
        float f = s_conf[i]; if (f != 0.0f) atomAddDev(&g_conf[i], f);
        float a = s_acc[i];  if (a != 0.0f) atomAddDev(&g_acc[i],  a);
    }
    if (tid == 0) atomicMax(g_maxlab, s_maxlab);
}

// ---------------- kernel 2: finalize scalar ----------------
__global__ void ece_finalize(const float* __restrict__ g_cnt,
                             const float* __restrict__ g_conf,
                             const float* __restrict__ g_acc,
                             const int*   __restrict__ g_maxlab,
                             float* __restrict__ out,
                             float invN)
{
    __shared__ float red[128];
    const int c = threadIdx.x;             // one class per thread
    const int num_classes = *g_maxlab + 1;

    float sce = 0.0f;
    if (c < CCLS && c < num_classes) {
        #pragma unroll
        for (int b = 0; b < NBINS; ++b) {
            const float cnt = g_cnt[c * NBINS + b];
            if (cnt > 0.0f) {
                const float gap = fabsf((g_conf[c * NBINS + b] - g_acc[c * NBINS + b]) / cnt);
                sce += gap * cnt * invN;
            }
        }
    }
    red[c] = sce;
    __syncthreads();
    #pragma unroll
    for (int off = 64; off >= 1; off >>= 1) {
        if (c < off) red[c] += red[c + off];
        __syncthreads();
    }
    if (c == 0) *out = red[0] / (float)num_classes;
}

// ---------------- host entry ----------------
extern "C" void kernel_launch(void* const* d_in, const int* in_sizes, int n_in,
                              void* d_out, int out_size, void* d_ws, size_t ws_size,
                              hipStream_t stream) {
    const float* logits = (const float*)d_in[0];
    const int*   labels = (const int*)d_in[1];
    const int N = in_sizes[1];             // labels count = rows

    float* ws     = (float*)d_ws;
    float* g_cnt  = ws;
    float* g_conf = ws + NSEG;
    float* g_acc  = ws + 2 * NSEG;
    int*   g_max  = (int*)(ws + 3 * NSEG);

    ece_zero_ws<<<(3 * NSEG + 1 + 255) / 256, 256, 0, stream>>>(ws);
    ece_accum<<<GRID_BLOCKS, BLOCK_THREADS, 0, stream>>>(
        logits, labels, g_cnt, g_conf, g_acc, g_max, N);
    ece_finalize<<<1, 128, 0, stream>>>(
        g_cnt, g_conf, g_acc, g_max, (float*)d_out, 1.0f / (float)N);
}